// cls_68015102099878
// MI455X (gfx1250) — compile-verified
//
#include <hip/hip_runtime.h>

typedef __attribute__((ext_vector_type(2))) float v2f;
typedef __attribute__((ext_vector_type(8))) float v8f;

#define B_ 8
#define L_ 4096
#define D_ 768
#define S_ 32
#define T_ 64
#define G_ 16

__global__ __launch_bounds__(256) void span_tem_wmma_kernel(
    const float* __restrict__ sen,        // [B,L,D]
    const float* __restrict__ arg,        // [B,S,T,D]
    const float* __restrict__ mask,       // [B,S,G]
    const int*   __restrict__ span_start, // [B,S]
    const int*   __restrict__ span_len,   // [B,S]
    const int*   __restrict__ offset_p,   // scalar
    float*       __restrict__ out)        // [B,S,G]
{
    __shared__ __align__(16) float span_lds[D_];        // 3 KB
    __shared__ __align__(16) float acc_lds[8 * 256];    // 8 KB: 8 waves x 8 vgprs x 32 lanes
    __shared__ float diag_lds[T_];

    const int bs  = blockIdx.x;          // b*S + s
    const int b   = bs / S_;
    const int tid = threadIdx.x;
    const int off = offset_p[0];

    // ---------- phase 1: span mean-pool into LDS (coalesced row streaming) ----------
    const int st = span_start[bs];
    const int ln = span_len[bs];
    int lo = st - off;           if (lo < 0)  lo = 0;
    int hi = st + ln + 1 + off;  if (hi > L_) hi = L_;
    float cnt = (float)(hi - lo); if (cnt < 1.0f) cnt = 1.0f;

    float a0 = 0.f, a1 = 0.f, a2 = 0.f;
    const float* srow = sen + (size_t)b * L_ * D_;
    for (int r = lo; r < hi; ++r) {
        const float* p = srow + (size_t)r * D_;
        a0 += p[tid];
        a1 += p[tid + 256];
        a2 += p[tid + 512];
    }
    const float inv = 1.0f / cnt;
    span_lds[tid]       = a0 * inv;
    span_lds[tid + 256] = a1 * inv;
    span_lds[tid + 512] = a2 * inv;
    __syncthreads();

    // ---------- phase 2: ssq[t] = diag(diff * diff^T) via V_WMMA_F32_16X16X4_F32 ----------
    // wave w: template tile (w&3), K-half (w>>2). A and B operands are the SAME
    // float2 per lane (symmetric tile): lane<16 -> (d=k,k+1) of row lane&15,
    // lane>=16 -> (d=k+2,k+3) of row lane&15.
    const int lane  = tid & 31;
    const int wave  = tid >> 5;
    const int tile  = wave & 3;
    const int khalf = wave >> 2;
    const int k0    = khalf * (D_ / 2);
    const int hiOff = (lane < 16) ? 0 : 2;
    const int tl    = lane & 15;

    const float* ap = arg + ((size_t)bs * T_ + (size_t)(tile * 16 + tl)) * D_ + (k0 + hiOff);
    const float* sp = span_lds + (k0 + hiOff);

    v8f c = {};
    #pragma unroll 4
    for (int k = 0; k < D_ / 2; k += 4) {
        v2f av = *(const v2f*)(ap + k);   // global_load_b64 (arg tile, 16B/row/step)
        v2f sv = *(const v2f*)(sp + k);   // ds_load_b64 (span broadcast per half-wave)
        v2f dv = av - sv;
        // D = A(16x4) * B(4x16) + C ; diagonal accumulates sum of squares
        c = __builtin_amdgcn_wmma_f32_16x16x4_f32(
                /*neg_a=*/false, dv, /*neg_b=*/false, dv,
                /*c_mod=*/(short)0, c, /*reuse_a=*/false, /*reuse_b=*/false);
    }

    // spill accumulator tiles to LDS
    #pragma unroll
    for (int r = 0; r < 8; ++r)
        acc_lds[wave * 256 + r * 32 + lane] = c[r];
    __syncthreads();

    // ---------- phase 3: extract diagonal, combine K-halves ----------
    if (tid < T_) {
        const int t  = tid;
        const int ti = t >> 4;   // template tile
        const int i  = t & 15;   // index within 16x16 C/D tile
        // C/D layout: i<8 -> vgpr i, lane i ; i>=8 -> vgpr i-8, lane i+16
        const int rr  = (i < 8) ? i : (i - 8);
        const int lc  = (i < 8) ? i : (i + 16);
        const int idx = rr * 32 + lc;
        diag_lds[t] = acc_lds[ti * 256 + idx] + acc_lds[(ti + 4) * 256 + idx];
    }
    __syncthreads();

    // ---------- phase 4: group max of logits = -(group min of ssq), apply mask ----------
    if (tid < G_) {
        const int g = tid;
        const float m0 = diag_lds[4 * g + 0];
        const float m1 = diag_lds[4 * g + 1];
        const float m2 = diag_lds[4 * g + 2];
        const float m3 = diag_lds[4 * g + 3];
        const float mn = fminf(fminf(m0, m1), fminf(m2, m3));
        out[(size_t)bs * G_ + g] = (-mn) * mask[(size_t)bs * G_ + g];
    }
}

extern "C" void kernel_launch(void* const* d_in, const int* in_sizes, int n_in,
                              void* d_out, int out_size, void* d_ws, size_t ws_size,
                              hipStream_t stream) {
    const float* sen        = (const float*)d_in[0]; // [B,L,D]
    const float* arg        = (const float*)d_in[1]; // [B,S,T,D]
    const float* mask       = (const float*)d_in[2]; // [B,S,G]
    const int*   span_start = (const int*)d_in[3];   // [B,S]
    const int*   span_len   = (const int*)d_in[4];   // [B,S]
    const int*   offset_p   = (const int*)d_in[5];   // scalar
    float* out = (float*)d_out;                      // [B,S,G]

    (void)in_sizes; (void)n_in; (void)out_size; (void)d_ws; (void)ws_size;

    dim3 grid(B_ * S_);   // 256 blocks, one per (b,s)
    dim3 block(256);      // 8 wave32 waves
    span_tem_wmma_kernel<<<grid, block, 0, stream>>>(
        sen, arg, mask, span_start, span_len, offset_p, out);
}